// Distance_16844861735173
// MI455X (gfx1250) — compile-verified
//
#include <hip/hip_runtime.h>

typedef float v2f __attribute__((ext_vector_type(2)));
typedef float v8f __attribute__((ext_vector_type(8)));

#define NA   2048
#define NB   (NA / 16)            // 128 atom blocks of 16
#define NT   (NB * (NB + 1) / 2)  // 8256 lower-triangle tile pairs
#define WPB  4                    // waves per block (128 threads)
#define LDP  17                   // padded LDS stride: conflict-free transpose
#define TF   (16 * LDP)

static_assert(NT % WPB == 0, "grid sized exactly; no partial waves");

struct WaveSmem {
  float d1[TF], x1[TF], y1[TF], z1[TF];   // forward tile:  d[i,j], vec[i,j]
  float d2[TF], x2[TF], y2[TF], z2[TF];   // backward tile: d[j,i], vec[j,i]
};

__device__ __forceinline__ v8f wmma16x16x4(v2f a, v2f b, v8f c) {
  // 8-arg VOP3P form: (neg_a, A, neg_b, B, c_mod, C, reuse_a, reuse_b)
  return __builtin_amdgcn_wmma_f32_16x16x4_f32(false, a, false, b, (short)0, c,
                                               false, false);
}

// One direction of a 16x16 tile: rows = R-block atoms, cols = C-block atoms.
// WMMA screens the 3 lattice-shifted images via
//   d_k^2 = |diff|^2 (exact, fed through accumulator C) + |L_k|^2 + 2 r_c.L_k - 2 r_r.L_k
// and the winning image is recomputed exactly with VALU so the stored d/vec
// match the reference arithmetic.
__device__ __forceinline__ void tile_direction(
    int h, int ln,
    float rRx, float rRy, float rRz,     // lane's row-block atom (row = ln)
    float rCx, float rCy, float rCz,     // lane's col-block atom (col = ln)
    const float (&LC)[3][3],             // lattice rows of the col atom
    const float* __restrict__ mask, int R0, int C0,
    float* __restrict__ dA, float* __restrict__ xA,
    float* __restrict__ yA, float* __restrict__ zA)
{
  // Issue the 8 batch_mask loads first: they are the only latency-exposed
  // global reads in the hot path; consuming them after the WMMAs hides HBM
  // latency behind matrix/VALU work.
  float mk[8];
#pragma unroll
  for (int v = 0; v < 8; ++v)
    mk[v] = mask[(size_t)(R0 + v + 8 * h) * NA + (C0 + ln)];

  // A(16x4) f32 wave32 layout: VGPR v, lane-half h -> K = v + 2h, row M = ln.
  v2f A; A.x = h ? rRz : rRx; A.y = h ? 1.0f : rRy;

  float d0sq[8], dfx[8], dfy[8], dfz[8];
#pragma unroll
  for (int v = 0; v < 8; ++v) {
    const int m = v + 8 * h;                 // C/D row this lane owns for VGPR v
    const float rix = __shfl(rRx, m, 32);
    const float riy = __shfl(rRy, m, 32);
    const float riz = __shfl(rRz, m, 32);
    const float dx = rix - rCx, dy = riy - rCy, dz = riz - rCz;
    dfx[v] = dx; dfy[v] = dy; dfz[v] = dz;
    d0sq[v] = fmaf(dx, dx, fmaf(dy, dy, dz * dz));   // exact |diff|^2
  }
  v8f Cacc;
#pragma unroll
  for (int v = 0; v < 8; ++v) Cacc[v] = d0sq[v];

  v8f D[3];
#pragma unroll
  for (int k = 0; k < 3; ++k) {
    const float Lx = LC[k][0], Ly = LC[k][1], Lz = LC[k][2];
    const float cterm = fmaf(Lx, Lx, fmaf(Ly, Ly, Lz * Lz)) +
                        2.0f * fmaf(rCx, Lx, fmaf(rCy, Ly, rCz * Lz));
    // B(4x16) col = (-2Lx, -2Ly, -2Lz, |L|^2 + 2 r_c.L); same K = v + 2h layout.
    v2f B; B.x = h ? (-2.0f * Lz) : (-2.0f * Lx);
    B.y = h ? cterm : (-2.0f * Ly);
    D[k] = wmma16x16x4(A, B, Cacc);          // v_wmma_f32_16x16x4_f32
  }

#pragma unroll
  for (int v = 0; v < 8; ++v) {
    const int m = v + 8 * h;
    const float mv = mk[v];
    // Branchless argmin (first-min tie rule via strict <): pure cndmask chains,
    // no exec-mask divergence around the selection.
    float best = mv * mv * d0sq[v];          // masked image-0 candidate
    float bx = mv * dfx[v], by = mv * dfy[v], bz = mv * dfz[v];
#pragma unroll
    for (int k = 0; k < 3; ++k) {
      const float dk = D[k][v];
      const float cx = dfx[v] - LC[k][0];
      const float cy = dfy[v] - LC[k][1];
      const float cz = dfz[v] - LC[k][2];
      const bool better = (dk < best);
      best = better ? dk : best;
      bx   = better ? cx : bx;
      by   = better ? cy : by;
      bz   = better ? cz : bz;
    }
    const float dsq = fmaf(bx, bx, fmaf(by, by, bz * bz));  // exact recompute
    const float dd  = sqrtf(dsq);
    const int idx = m * LDP + ln;
    dA[idx] = dd; xA[idx] = bx; yA[idx] = by; zA[idx] = bz;
  }
}

__global__ __launch_bounds__(WPB * 32)
void Distance_minimage_kernel(const float* __restrict__ r,
                              const float* __restrict__ mask,
                              const float* __restrict__ lat,
                              float* __restrict__ dout,
                              float* __restrict__ vout)
{
  __shared__ WaveSmem smem[WPB];
  const int lane = threadIdx.x & 31;
  const int wv   = threadIdx.x >> 5;
  const int t    = blockIdx.x * WPB + wv;    // grid sized exactly: t < NT
  WaveSmem& S = smem[wv];

  // t -> (bi >= bj) via triangular-number inversion
  int bi = (int)((sqrtf(8.0f * (float)t + 1.0f) - 1.0f) * 0.5f);
  while ((bi + 1) * (bi + 2) / 2 <= t) ++bi;
  while (bi * (bi + 1) / 2 > t) --bi;
  const int bj = t - bi * (bi + 1) / 2;
  const int I0 = bi * 16, J0 = bj * 16;
  const bool diag = (bi == bj);

  const int h = lane >> 4, ln = lane & 15;
  const int ai = I0 + ln, aj = J0 + ln;

  const float rIx = r[3 * ai + 0], rIy = r[3 * ai + 1], rIz = r[3 * ai + 2];
  const float rJx = r[3 * aj + 0], rJy = r[3 * aj + 1], rJz = r[3 * aj + 2];
  float LI[3][3], LJ[3][3];
#pragma unroll
  for (int k = 0; k < 3; ++k)
#pragma unroll
    for (int c = 0; c < 3; ++c) {
      LI[k][c] = lat[9 * ai + 3 * k + c];
      LJ[k][c] = lat[9 * aj + 3 * k + c];
    }

  // d[i,j]: rows I, cols J, images from lattice[j]; d[j,i]: roles swapped.
  tile_direction(h, ln, rIx, rIy, rIz, rJx, rJy, rJz, LJ, mask, I0, J0,
                 S.d1, S.x1, S.y1, S.z1);
  tile_direction(h, ln, rJx, rJy, rJz, rIx, rIy, rIz, LI, mask, J0, I0,
                 S.d2, S.x2, S.y2, S.z2);
  __syncthreads();

  const int ncol = NA - 1;
#pragma unroll
  for (int v = 0; v < 8; ++v) {
    const int m = v + 8 * h, n = ln;
    const int fwd = m * LDP + n, rev = n * LDP + m;

    // lower-triangle position (i = I0+m, j = J0+n), i > j
    if (!diag || m > n) {
      const float a = S.d1[fwd], b = S.d2[rev];   // d[i,j], d[j,i]
      const bool useA = (a <= b);                 // reference key2 tie -> tril
      const float dmin = useA ? a : b;
      const float sx = useA ? S.x1[fwd] : S.x2[rev];
      const float sy = useA ? S.y1[fwd] : S.y2[rev];
      const float sz = useA ? S.z1[fwd] : S.z2[rev];
      const size_t o = (size_t)(I0 + m) * ncol + (J0 + n);  // j < i -> oc = j
      const bool z = (dmin == 0.0f);
      dout[o] = dmin;
      vout[3 * o + 0] = z ? 0.0f : sx / dmin;
      vout[3 * o + 1] = z ? 0.0f : sy / dmin;
      vout[3 * o + 2] = z ? 0.0f : sz / dmin;
    }

    // upper-triangle position (i = J0+m, j = I0+n), i < j: value = -sel(j,i)
    if (!diag || m < n) {
      const float a = S.d2[fwd];   // d[i,j]
      const float b = S.d1[rev];   // d[j,i]
      const bool useB = (b <= a);  // lower partner keeps its own on ties
      const float dmin = useB ? b : a;
      const float sx = useB ? S.x1[rev] : S.x2[fwd];
      const float sy = useB ? S.y1[rev] : S.y2[fwd];
      const float sz = useB ? S.z1[rev] : S.z2[fwd];
      const size_t o = (size_t)(J0 + m) * ncol + (I0 + n - 1); // j > i -> oc = j-1
      const bool z = (dmin == 0.0f);
      dout[o] = dmin;
      vout[3 * o + 0] = z ? 0.0f : -sx / dmin;
      vout[3 * o + 1] = z ? 0.0f : -sy / dmin;
      vout[3 * o + 2] = z ? 0.0f : -sz / dmin;
    }
  }
}

extern "C" void kernel_launch(void* const* d_in, const int* in_sizes, int n_in,
                              void* d_out, int out_size, void* d_ws, size_t ws_size,
                              hipStream_t stream) {
  (void)in_sizes; (void)n_in; (void)out_size; (void)d_ws; (void)ws_size;
  const float* r    = (const float*)d_in[0];   // (1, NA, 3)
  const float* mask = (const float*)d_in[1];   // (1, NA, NA, 1)
  // d_in[2] = loop_mask (bool): unused by the reference math
  const float* lat  = (const float*)d_in[3];   // (1, NA, 3, 3)
  float* dout = (float*)d_out;                  // d: NA x (NA-1)
  float* vout = dout + (size_t)NA * (NA - 1);   // vec_norm: NA x (NA-1) x 3
  dim3 grid(NT / WPB), block(WPB * 32);
  Distance_minimage_kernel<<<grid, block, 0, stream>>>(r, mask, lat, dout, vout);
}